// Aggregator_89043261981078
// MI455X (gfx1250) — compile-verified
//
#include <hip/hip_runtime.h>

#define N_NODES 50000
#define N_EDGES 800000
#define DIM 64

typedef __attribute__((ext_vector_type(2))) float v2f;
typedef __attribute__((ext_vector_type(8))) float v8f;
typedef __attribute__((address_space(3))) float lds_f32;

// ---------------------------------------------------------------------------
// Kernel 0: zero the N_h accumulator in workspace (harness poisons ws).
// ---------------------------------------------------------------------------
__global__ void zero_nh_kernel(float4* __restrict__ nh4, int n4) {
    int i = blockIdx.x * blockDim.x + threadIdx.x;
    if (i < n4) nh4[i] = make_float4(0.f, 0.f, 0.f, 0.f);
}

// ---------------------------------------------------------------------------
// Kernel 1: edge gather + scatter-add.
//   msg = embed[src] * out_sqrt_degree[src] * edge_weight
//   N_h[dst] += msg        (f32 atomics; N_h is 12.8 MB -> L2 resident)
// 16 threads per edge, each thread handles 4 consecutive floats (float4 gather).
// ---------------------------------------------------------------------------
__global__ __launch_bounds__(256) void edge_scatter_kernel(
    const float* __restrict__ embed,
    const int*   __restrict__ src,
    const int*   __restrict__ dst,
    const float* __restrict__ ew,
    const float* __restrict__ outdeg,
    float*       __restrict__ nh) {
    int tid = blockIdx.x * blockDim.x + threadIdx.x;
    int e = tid >> 4;
    if (e >= N_EDGES) return;
    int q = (tid & 15) << 2;                 // element offset 0..60

    int s = src[e];
    int d = dst[e];
    float w = ew[e] * outdeg[s];

    const float4* xr = (const float4*)(embed + (long)s * DIM + q);
    float4 v = *xr;
    float* o = nh + (long)d * DIM + q;
    __hip_atomic_fetch_add(o + 0, v.x * w, __ATOMIC_RELAXED, __HIP_MEMORY_SCOPE_AGENT);
    __hip_atomic_fetch_add(o + 1, v.y * w, __ATOMIC_RELAXED, __HIP_MEMORY_SCOPE_AGENT);
    __hip_atomic_fetch_add(o + 2, v.z * w, __ATOMIC_RELAXED, __HIP_MEMORY_SCOPE_AGENT);
    __hip_atomic_fetch_add(o + 3, v.w * w, __ATOMIC_RELAXED, __HIP_MEMORY_SCOPE_AGENT);
}

// ---------------------------------------------------------------------------
// Kernel 2: fused  out = leaky_relu((embed + N_h * in_deg) @ W^T + b)
// using V_WMMA_F32_16X16X4_F32 (exact f32 path).
//
// W (16 KB) is staged into LDS ONCE per block via the gfx1250 async
// global->LDS copy (global_load_async_to_lds_b128, ASYNCcnt), then all B
// fragments come from ds_load_b64. Rows padded to 68 floats so a B-fragment
// read (lane i: banks 4i+k..4i+k+1, lane 16+i: 4i+k+2..4i+k+3) covers all 64
// LDS banks exactly once -> conflict-free.
//
// Block = 128 threads = 4 waves; each wave owns a 16-row tile -> 64 rows/block.
// Per wave: acc[4] (N-tiles of 16 cols), 16 K-steps of 4 -> 64 WMMAs.
//
// Fragment layouts (ISA 7.12.2, wave32):
//   A 16x4 f32 (2 VGPRs): lanes 0-15: M=lane, K=k0+{0,1}; lanes 16-31: K=k0+{2,3}
//   B 4x16 f32 (2 VGPRs): lanes 0-15: N=lane, K=k0+{0,1}; lanes 16-31: K=k0+{2,3}
//     B[k][n] = W^T[k][n] = W[n][k]  -> contiguous float2 from row n of W
//   C/D 16x16 f32 (8 VGPRs): VGPR r: M = r + 8*(lane>=16), N = lane&15
// ---------------------------------------------------------------------------
__global__ __launch_bounds__(128) void gcn_gemm_wmma_kernel(
    const float* __restrict__ embed,
    const float* __restrict__ nh,
    const float* __restrict__ indeg,
    const float* __restrict__ W,      // [DIM, DIM] row-major
    const float* __restrict__ bias,   // [DIM]
    float*       __restrict__ out) {
    __shared__ float sW[64][68];      // padded: bank(n,k) = (4n + k) & 63

    const int lane = threadIdx.x & 31;
    const int wave = threadIdx.x >> 5;
    const int half = lane >> 4;       // 0: lanes 0-15, 1: lanes 16-31
    const int l15  = lane & 15;

    // ---- async-stage W into LDS: 1024 b128 chunks, 8 per thread ----
    #pragma unroll
    for (int i = 0; i < 8; ++i) {
        int c  = threadIdx.x + i * 128;   // chunk id 0..1023
        int n  = c >> 4;                  // W row
        int c4 = (c & 15) << 2;           // starting float within row
        const float* gp = W + (long)n * DIM + c4;
        lds_f32* lp = (lds_f32*)&sW[n][c4];
        asm volatile("global_load_async_to_lds_b128 %0, %1, off"
                     :: "v"(lp), "v"(gp) : "memory");
    }
    asm volatile("s_wait_asynccnt 0" ::: "memory");
    __syncthreads();

    const long row0 = (long)blockIdx.x * 64 + (long)wave * 16;

    // Row this lane's A-fragment reads (clamped so EXEC stays all-1s for WMMA).
    long arow = row0 + l15;
    if (arow >= N_NODES) arow = N_NODES - 1;
    const float insd = indeg[arow];
    const float* xe = embed + arow * DIM;
    const float* xn = nh    + arow * DIM;

    v8f acc0 = {}, acc1 = {}, acc2 = {}, acc3 = {};

    #pragma unroll
    for (int kk = 0; kk < 16; ++kk) {
        const int k = kk * 4 + half * 2;     // this lane's K pair within the K-step
        v2f a;
        a.x = xe[k]     + xn[k]     * insd;  // X = embed + N_h * in_sqrt_degree
        a.y = xe[k + 1] + xn[k + 1] * insd;

        v2f b0 = *(const v2f*)&sW[l15 +  0][k];
        v2f b1 = *(const v2f*)&sW[l15 + 16][k];
        v2f b2 = *(const v2f*)&sW[l15 + 32][k];
        v2f b3 = *(const v2f*)&sW[l15 + 48][k];

        acc0 = __builtin_amdgcn_wmma_f32_16x16x4_f32(false, a, false, b0, (short)0, acc0, false, false);
        acc1 = __builtin_amdgcn_wmma_f32_16x16x4_f32(false, a, false, b1, (short)0, acc1, false, false);
        acc2 = __builtin_amdgcn_wmma_f32_16x16x4_f32(false, a, false, b2, (short)0, acc2, false, false);
        acc3 = __builtin_amdgcn_wmma_f32_16x16x4_f32(false, a, false, b3, (short)0, acc3, false, false);
    }

    // Epilogue: bias + LeakyReLU(0.01), guarded stores for the tail tile.
    v8f accs[4] = {acc0, acc1, acc2, acc3};
    #pragma unroll
    for (int t = 0; t < 4; ++t) {
        const int n = t * 16 + l15;
        const float bn = bias[n];
        #pragma unroll
        for (int r = 0; r < 8; ++r) {
            const long m = row0 + r + 8 * half;
            if (m < N_NODES) {
                float v = accs[t][r] + bn;
                out[m * DIM + n] = v > 0.f ? v : 0.01f * v;
            }
        }
    }
}

// ---------------------------------------------------------------------------
extern "C" void kernel_launch(void* const* d_in, const int* in_sizes, int n_in,
                              void* d_out, int out_size, void* d_ws, size_t ws_size,
                              hipStream_t stream) {
    const float* embed  = (const float*)d_in[0];   // [N, D]
    const int*   src    = (const int*)  d_in[1];   // [E]
    const int*   dst    = (const int*)  d_in[2];   // [E]
    const float* ew     = (const float*)d_in[3];   // [E, 1]
    const float* outdeg = (const float*)d_in[4];   // [N, 1]
    const float* indeg  = (const float*)d_in[5];   // [N, 1]
    const float* W      = (const float*)d_in[6];   // [D, D]
    const float* bias   = (const float*)d_in[7];   // [D]
    float*       out    = (float*)d_out;           // [N, D]
    float*       nh     = (float*)d_ws;            // [N, D] scratch accumulator

    // 1) zero N_h (N*D/4 float4 elements)
    {
        int n4 = N_NODES * DIM / 4;                // 800,000
        int blocks = (n4 + 255) / 256;
        zero_nh_kernel<<<blocks, 256, 0, stream>>>((float4*)nh, n4);
    }
    // 2) edge scatter-add (16 threads/edge)
    {
        long threads = (long)N_EDGES * 16;
        int blocks = (int)((threads + 255) / 256); // 50,000
        edge_scatter_kernel<<<blocks, 256, 0, stream>>>(embed, src, dst, ew, outdeg, nh);
    }
    // 3) fused scale + add + GEMM(W^T) + bias + LeakyReLU via f32 WMMA
    {
        int blocks = (N_NODES + 63) / 64;          // 782
        gcn_gemm_wmma_kernel<<<blocks, 128, 0, stream>>>(embed, nh, indeg, W, bias, out);
    }
}